// GNSSolver_20873541059066
// MI455X (gfx1250) — compile-verified
//
#include <hip/hip_runtime.h>
#include <cstdint>
#include <cstddef>

#define BB    8
#define NN    2048
#define DD    100
#define KK    10
#define KIN   102      // D+2 input features to leap MLP
#define KPAD  104      // K padded to multiple of 4
#define KSTR  106      // LDS K stride (pad to de-conflict banks: gcd-free vs 64)
#define NPAD  112      // output cols padded to multiple of 16
#define ROWS  8        // Y rows per workgroup in matvec

typedef __attribute__((ext_vector_type(2))) float v2f;
typedef __attribute__((ext_vector_type(8))) float v8f;

// ---------------------------------------------------------------------------
// Setup: connectivity bitmask, reciprocal degree, complete-row flag
// ---------------------------------------------------------------------------
__global__ void k_conn(const float* __restrict__ Yr, const float* __restrict__ Yi,
                       unsigned* __restrict__ connW, float* __restrict__ rdeg,
                       int* __restrict__ full) {
  __shared__ unsigned w[64];
  __shared__ int degS;
  int i = blockIdx.x, b = blockIdx.y, t = threadIdx.x;
  if (t < 64) w[t] = 0u;
  if (t == 0) degS = 0;
  __syncthreads();
  size_t base = ((size_t)(b * NN + i)) * NN;
  unsigned m = 0;
  int j0 = t * 8;
#pragma unroll
  for (int u = 0; u < 8; ++u) {
    int j = j0 + u;
    float yr = Yr[base + j], yi = Yi[base + j];
    int nz = (int)((yr != 0.f) | (yi != 0.f)) & (int)(j != i);
    m |= ((unsigned)nz) << u;
  }
  atomicOr(&w[t >> 2], m << ((t & 3) * 8));
  __syncthreads();
  if (t < 64) {
    connW[((size_t)(b * NN + i)) * 64 + t] = w[t];
    atomicAdd(&degS, __popc(w[t]));
  }
  __syncthreads();
  if (t == 0) {
    int deg = degS;
    rdeg[b * NN + i] = 1.0f / (float)(deg > 1 ? deg : 1);
    full[b * NN + i] = (deg == NN - 1) ? 1 : 0;
  }
}

// ---------------------------------------------------------------------------
// Embedding MLPs (2 -> D per bus type) + flat-start V init (into d_out)
// ---------------------------------------------------------------------------
__global__ void k_embed(const int* __restrict__ bt,
                        const float* __restrict__ P, const float* __restrict__ Q,
                        const float* __restrict__ Wsl, const float* __restrict__ bsl,
                        const float* __restrict__ Wg,  const float* __restrict__ bg,
                        const float* __restrict__ Wl,  const float* __restrict__ bl,
                        float* __restrict__ H, float* __restrict__ out) {
  int idx = blockIdx.x * blockDim.x + threadIdx.x;
  if (idx >= BB * NN) return;
  int ty = bt[idx];
  float vm = (ty == 1) ? 1.06f : 1.0f;
  out[idx * 2 + 0] = vm;
  out[idx * 2 + 1] = 0.f;
  float f0, f1;
  const float *W, *bb_;
  if (ty == 1)      { f0 = vm;     f1 = 0.f;    W = Wsl; bb_ = bsl; }
  else if (ty == 2) { f0 = P[idx]; f1 = vm;     W = Wg;  bb_ = bg; }
  else              { f0 = P[idx]; f1 = Q[idx]; W = Wl;  bb_ = bl; }
  for (int d = 0; d < DD; ++d)
    H[(size_t)idx * DD + d] = tanhf(f0 * W[d] + f1 * W[DD + d] + bb_[d]);
}

// ---------------------------------------------------------------------------
// c = Vmag*cos(Vang), s = Vmag*sin(Vang)
// ---------------------------------------------------------------------------
__global__ void k_cs(const float* __restrict__ out,
                     float* __restrict__ c, float* __restrict__ s) {
  int idx = blockIdx.x * blockDim.x + threadIdx.x;
  if (idx >= BB * NN) return;
  float vm = out[idx * 2 + 0], va = out[idx * 2 + 1];
  c[idx] = vm * cosf(va);
  s[idx] = vm * sinf(va);
}

// ---------------------------------------------------------------------------
// Bandwidth stage: Ir = Yr c - Yi s ; Ii = Yr s + Yi c   (streams Yr, Yi)
// ---------------------------------------------------------------------------
__global__ void __launch_bounds__(256)
k_matvec(const float* __restrict__ Yr, const float* __restrict__ Yi,
         const float* __restrict__ c, const float* __restrict__ s,
         float* __restrict__ Ir, float* __restrict__ Ii) {
  __shared__ float cs[NN];
  __shared__ float ss[NN];
  __shared__ float redr[256];
  __shared__ float redi[256];
  int b = blockIdx.y, t = threadIdx.x;
  for (int j = t; j < NN; j += 256) {
    cs[j] = c[b * NN + j];
    ss[j] = s[b * NN + j];
  }
  __syncthreads();
  for (int r = 0; r < ROWS; ++r) {
    int i = blockIdx.x * ROWS + r;
    size_t base = ((size_t)(b * NN + i)) * NN;
    if (r + 1 < ROWS) {  // hint the next row into cache (global_prefetch_b8)
      __builtin_prefetch(Yr + base + NN + t * 8, 0, 1);
      __builtin_prefetch(Yi + base + NN + t * 8, 0, 1);
    }
    const float4* yr4 = (const float4*)(Yr + base);
    const float4* yi4 = (const float4*)(Yi + base);
    float ir = 0.f, ii = 0.f;
#pragma unroll
    for (int q = 0; q < 2; ++q) {
      int f = t * 2 + q;
      float4 a = yr4[f];
      float4 d = yi4[f];
      int j = f * 4;
      ir += a.x * cs[j + 0] - d.x * ss[j + 0];  ii += a.x * ss[j + 0] + d.x * cs[j + 0];
      ir += a.y * cs[j + 1] - d.y * ss[j + 1];  ii += a.y * ss[j + 1] + d.y * cs[j + 1];
      ir += a.z * cs[j + 2] - d.z * ss[j + 2];  ii += a.z * ss[j + 2] + d.z * cs[j + 2];
      ir += a.w * cs[j + 3] - d.w * ss[j + 3];  ii += a.w * ss[j + 3] + d.w * cs[j + 3];
    }
    redr[t] = ir; redi[t] = ii;
    __syncthreads();
    for (int st = 128; st > 0; st >>= 1) {
      if (t < st) { redr[t] += redr[t + st]; redi[t] += redi[t + st]; }
      __syncthreads();
    }
    if (t == 0) { Ir[b * NN + i] = redr[0]; Ii[b * NN + i] = redi[0]; }
    __syncthreads();
  }
}

// ---------------------------------------------------------------------------
// Deterministic two-stage column sum:  S[b,d] = sum_i H[b,i,d]
// ---------------------------------------------------------------------------
__global__ void k_colsum1(const float* __restrict__ H, float* __restrict__ part) {
  int cb = blockIdx.x, b = blockIdx.y, t = threadIdx.x;  // 128 threads
  int i0 = cb * (NN / 16);
  for (int d = t; d < DD; d += 128) {
    float acc = 0.f;
    for (int i = i0; i < i0 + NN / 16; ++i)
      acc += H[((size_t)(b * NN + i)) * DD + d];
    part[(b * 16 + cb) * DD + d] = acc;
  }
}

__global__ void k_colsum2(const float* __restrict__ part, float* __restrict__ S) {
  int b = blockIdx.x, t = threadIdx.x;  // 128 threads
  for (int d = t; d < DD; d += 128) {
    float acc = 0.f;
    for (int cb = 0; cb < 16; ++cb) acc += part[(b * 16 + cb) * DD + d];
    S[b * DD + d] = acc;
  }
}

// ---------------------------------------------------------------------------
// Power mismatches dP, dQ
// ---------------------------------------------------------------------------
__global__ void k_mismatch(const int* __restrict__ bt,
                           const float* __restrict__ c, const float* __restrict__ s,
                           const float* __restrict__ Ir, const float* __restrict__ Ii,
                           const float* __restrict__ P, const float* __restrict__ Q,
                           float* __restrict__ dP, float* __restrict__ dQ) {
  int idx = blockIdx.x * blockDim.x + threadIdx.x;
  if (idx >= BB * NN) return;
  float cc = c[idx], ssv = s[idx], ir = Ir[idx], ii = Ii[idx];
  float Pc = cc * ir + ssv * ii;
  float Qc = ssv * ir - cc * ii;
  int ty = bt[idx];
  dP[idx] = (ty == 1) ? 0.f : (P[idx] - Pc);
  dQ[idx] = (ty == 3) ? (Q[idx] - Qc) : 0.f;
}

// ---------------------------------------------------------------------------
// Hn = A @ H : exact complete-graph fast path, exact bitmask fallback
// ---------------------------------------------------------------------------
__global__ void k_hn(const float* __restrict__ H, const float* __restrict__ S,
                     const float* __restrict__ rdeg, const int* __restrict__ full,
                     const unsigned* __restrict__ connW, float* __restrict__ Hn) {
  int idx = blockIdx.x * blockDim.x + threadIdx.x;  // (b*NN+i)*DD + d
  if (idx >= BB * NN * DD) return;
  int d = idx % DD;
  int bi = idx / DD;
  int b = bi / NN;
  float acc;
  if (full[bi]) {
    acc = S[b * DD + d] - H[idx];
  } else {
    acc = 0.f;
    const unsigned* w = connW + (size_t)bi * 64;
    for (int wd = 0; wd < 64; ++wd) {
      unsigned m = w[wd];
      while (m) {
        int bit = __ffs(m) - 1;
        m &= m - 1;
        int j = wd * 32 + bit;
        acc += H[((size_t)(b * NN + j)) * DD + d];
      }
    }
  }
  Hn[idx] = acc * rdeg[bi];
}

// ---------------------------------------------------------------------------
// WMMA GEMM:  H += tanh([Hn|dP|dQ] @ Wk + bk)   using V_WMMA_F32_16X16X4_F32
// Wk is staged TRANSPOSED in LDS (WkT[col][k], K-stride 106 words) so each
// lane's B fragment (two consecutive K values of one column) is a single
// aligned ds_load_b64 straight into an even VGPR pair. A fragments are built
// as v2f pairs so they feed WMMA without repacking movs.
// ---------------------------------------------------------------------------
__global__ void __launch_bounds__(256)
k_gemm(const float* __restrict__ Hn, const float* __restrict__ dP,
       const float* __restrict__ dQ, const float* __restrict__ Wk,
       const float* __restrict__ bk, float* __restrict__ H) {
  __shared__ float WkT[NPAD * KSTR];  // [col][k], padded stride
  __shared__ float biasL[NPAD];
  int t = threadIdx.x;
  for (int idx = t; idx < NPAD * KSTR; idx += 256) {
    int d = idx / KSTR, k = idx % KSTR;
    WkT[idx] = (k < KIN && d < DD) ? Wk[k * DD + d] : 0.f;
  }
  if (t < NPAD) biasL[t] = (t < DD) ? bk[t] : 0.f;
  __syncthreads();

  int wave = t >> 5, lane = t & 31;
  int half = lane >> 4, m = lane & 15;
  int row0 = (blockIdx.x * 8 + wave) * 16;
  int gi = row0 + m;  // this lane's A row (flattened b*N+i)

  // A fragments: lane holds A[m][k], k = 4*tt + 2*half + v  (ISA 16x4 layout)
  v2f aP[26];
#pragma unroll
  for (int tt = 0; tt < 26; ++tt) {
#pragma unroll
    for (int v = 0; v < 2; ++v) {
      int k = 4 * tt + 2 * half + v;
      float x = (k < DD) ? Hn[(size_t)gi * DD + k]
              : (k == DD) ? dP[gi]
              : (k == DD + 1) ? dQ[gi]
              : 0.f;
      aP[tt][v] = x;
    }
  }

  v8f acc[7];
#pragma unroll
  for (int n = 0; n < 7; ++n) {
    v8f z = {0.f, 0.f, 0.f, 0.f, 0.f, 0.f, 0.f, 0.f};
    acc[n] = z;
  }

#pragma unroll
  for (int tt = 0; tt < 26; ++tt) {
    v2f av = aP[tt];
    int kb = 4 * tt + 2 * half;  // 8-byte aligned k-pair start
#pragma unroll
    for (int n = 0; n < 7; ++n) {
      v2f bv = *(const v2f*)&WkT[(n * 16 + m) * KSTR + kb];  // ds_load_b64
      acc[n] = __builtin_amdgcn_wmma_f32_16x16x4_f32(
          /*neg_a=*/false, av, /*neg_b=*/false, bv,
          /*c_mod=*/(short)0, acc[n], /*reuse_a=*/false, /*reuse_b=*/false);
    }
  }

  // Epilogue: C lane layout (row = v + 8*half, col = m within tile)
#pragma unroll
  for (int n = 0; n < 7; ++n) {
    int d = n * 16 + m;
    if (d >= DD) continue;
    float bias = biasL[d];
#pragma unroll
    for (int v = 0; v < 8; ++v) {
      int row = row0 + v + 8 * half;
      size_t p = (size_t)row * DD + d;
      H[p] = H[p] + tanhf(acc[n][v] + bias);
    }
  }
}

// ---------------------------------------------------------------------------
// Decode Newton step:  dV = H @ W_dec + b_dec ; masked V updates (into d_out)
// ---------------------------------------------------------------------------
__global__ void k_decode(const int* __restrict__ bt, const float* __restrict__ H,
                         const float* __restrict__ Wd, const float* __restrict__ bd,
                         float* __restrict__ out) {
  int idx = blockIdx.x * blockDim.x + threadIdx.x;
  if (idx >= BB * NN) return;
  float d0 = bd[0], d1 = bd[1];
  const float* h = H + (size_t)idx * DD;
  for (int d = 0; d < DD; ++d) {
    float hv = h[d];
    d0 += hv * Wd[d * 2 + 0];
    d1 += hv * Wd[d * 2 + 1];
  }
  int ty = bt[idx];
  if (ty == 3) out[idx * 2 + 0] += d0;  // only PQ buses update |V|
  if (ty != 1) out[idx * 2 + 1] += d1;  // slack angle fixed
}

// ---------------------------------------------------------------------------
extern "C" void kernel_launch(void* const* d_in, const int* in_sizes, int n_in,
                              void* d_out, int out_size, void* d_ws, size_t ws_size,
                              hipStream_t stream) {
  (void)in_sizes; (void)n_in; (void)out_size; (void)ws_size;
  const int*   bt    = (const int*)  d_in[0];
  const float* Yr    = (const float*)d_in[1];
  const float* Yi    = (const float*)d_in[2];
  const float* P     = (const float*)d_in[3];
  const float* Q     = (const float*)d_in[4];
  const float* Wsl   = (const float*)d_in[5];
  const float* bsl   = (const float*)d_in[6];
  const float* Wg    = (const float*)d_in[7];
  const float* bg    = (const float*)d_in[8];
  const float* Wl    = (const float*)d_in[9];
  const float* bl    = (const float*)d_in[10];
  const float* Wleap = (const float*)d_in[11];
  const float* bleap = (const float*)d_in[12];
  const float* Wd    = (const float*)d_in[13];
  const float* bd    = (const float*)d_in[14];
  float* out = (float*)d_out;

  char* ws = (char*)d_ws;
  size_t o = 0;
  auto alloc = [&](size_t bytes) -> char* {
    char* p = ws + o;
    o += (bytes + 255) & ~(size_t)255;
    return p;
  };
  unsigned* connW = (unsigned*)alloc((size_t)BB * NN * 64 * 4);
  float* rdeg = (float*)alloc((size_t)BB * NN * 4);
  int*   full = (int*)  alloc((size_t)BB * NN * 4);
  float* c    = (float*)alloc((size_t)BB * NN * 4);
  float* s    = (float*)alloc((size_t)BB * NN * 4);
  float* Ir   = (float*)alloc((size_t)BB * NN * 4);
  float* Ii   = (float*)alloc((size_t)BB * NN * 4);
  float* dP   = (float*)alloc((size_t)BB * NN * 4);
  float* dQ   = (float*)alloc((size_t)BB * NN * 4);
  float* S    = (float*)alloc((size_t)BB * DD * 4);
  float* part = (float*)alloc((size_t)BB * 16 * DD * 4);
  float* H    = (float*)alloc((size_t)BB * NN * DD * 4);
  float* Hn   = (float*)alloc((size_t)BB * NN * DD * 4);

  k_conn<<<dim3(NN, BB), 256, 0, stream>>>(Yr, Yi, connW, rdeg, full);
  k_embed<<<(BB * NN) / 256, 256, 0, stream>>>(bt, P, Q, Wsl, bsl, Wg, bg, Wl, bl, H, out);

  for (int k = 0; k < KK; ++k) {
    k_cs<<<(BB * NN) / 256, 256, 0, stream>>>(out, c, s);
    k_matvec<<<dim3(NN / ROWS, BB), 256, 0, stream>>>(Yr, Yi, c, s, Ir, Ii);
    k_colsum1<<<dim3(16, BB), 128, 0, stream>>>(H, part);
    k_colsum2<<<BB, 128, 0, stream>>>(part, S);
    k_mismatch<<<(BB * NN) / 256, 256, 0, stream>>>(bt, c, s, Ir, Ii, P, Q, dP, dQ);
    k_hn<<<(BB * NN * DD) / 256, 256, 0, stream>>>(H, S, rdeg, full, connW, Hn);
    k_gemm<<<(BB * NN / 16) / 8, 256, 0, stream>>>(
        Hn, dP, dQ, Wleap + (size_t)k * KIN * DD, bleap + (size_t)k * DD, H);
    k_decode<<<(BB * NN) / 256, 256, 0, stream>>>(bt, H, Wd, bd, out);
  }
}